// QLSTMPlus_65481071406418
// MI455X (gfx1250) — compile-verified
//
#include <hip/hip_runtime.h>
#include <math.h>

typedef __attribute__((ext_vector_type(2))) float v2f;
typedef __attribute__((ext_vector_type(8))) float v8f;

#define NQ 12
#define NSTATE 4096
#define NTHR 256
#define BATCH 64
#define SEQLEN 16
#define HID 4

// State index convention: qubit q <-> bit q of the linear amplitude index.
// (Different from JAX's internal ordering but physically identical; the
// measured Z expectations per qubit are invariant under this relabeling.)
__device__ __forceinline__ int addr_of(int g, int row, int col) {
  // Amplitude address for "group" view g: row = 4-bit group nibble,
  // col = remaining 8 bits.
  if (g == 0) return (col << 4) | row;                                  // bits 0..3
  if (g == 1) return ((col >> 4) << 8) | (row << 4) | (col & 15);       // bits 4..7
  return (row << 8) | col;                                              // bits 8..11
}

__global__ __launch_bounds__(NTHR) void qlstm_circuit_kernel(
    const float* __restrict__ inputs,
    const float* __restrict__ wF, const float* __restrict__ wI,
    const float* __restrict__ wU, const float* __restrict__ wO,
    const float* __restrict__ hbuf,
    float* __restrict__ zbuf, int t)
{
  __shared__ float sRe[NSTATE];            // 16 KB
  __shared__ float sIm[NSTATE];            // 16 KB
  __shared__ float sUr [3][16][16];        // fused 4-qubit gate, real
  __shared__ float sUi [3][16][16];        // imag
  __shared__ float sUiN[3][16][16];        // -imag (F32 WMMA has no A-neg)
  __shared__ float sG[12][2][2][2];        // per-qubit Rot gate [q][r][c][re/im]
  __shared__ float sMg[3][16];             // RX product-state magnitude tables
  __shared__ float sCos[NQ], sSin[NQ];
  __shared__ float sWred[8][NQ];

  const int tid  = threadIdx.x;
  const int b    = blockIdx.x;   // batch sample
  const int gate = blockIdx.y;   // 0=forget 1=input 2=update 3=output
  const float* W = (gate == 0) ? wF : ((gate == 1) ? wI : ((gate == 2) ? wU : wO));

  // ---- angles: comb = concat(x_t[b], h[b]) ----
  if (tid < NQ) {
    float ang = (tid < 8) ? inputs[(size_t)t * BATCH * 8 + b * 8 + tid]
                          : hbuf[b * HID + (tid - 8)];
    sCos[tid] = cosf(0.5f * ang);
    sSin[tid] = sinf(0.5f * ang);
  }
  __syncthreads();

  // ---- RX layer on |0..0> is a product state: amp(s) = M(s) * (-i)^popcount(s)
  if (tid < 48) {
    int g = tid >> 4, j = tid & 15;
    float m = 1.0f;
    #pragma unroll
    for (int bit = 0; bit < 4; ++bit) {
      int q = 4 * g + bit;
      m *= ((j >> bit) & 1) ? sSin[q] : sCos[q];
    }
    sMg[g][j] = m;
  }
  __syncthreads();
  for (int s = tid; s < NSTATE; s += NTHR) {
    float m = sMg[0][s & 15] * sMg[1][(s >> 4) & 15] * sMg[2][(s >> 8) & 15];
    int k = __popc(s) & 3;                  // (-i)^k cycle: 1, -i, -1, i
    sRe[s] = (k == 0) ? m : ((k == 2) ? -m : 0.0f);
    sIm[s] = (k == 1) ? -m : ((k == 3) ? m : 0.0f);
  }

  for (int l = 0; l < 2; ++l) {
    // ---- per-qubit Rot(phi,theta,omega) = RZ(om) RY(th) RZ(phi) ----
    __syncthreads();
    if (tid < NQ) {
      float phi = W[(l * NQ + tid) * 3 + 0];
      float th  = W[(l * NQ + tid) * 3 + 1];
      float om  = W[(l * NQ + tid) * 3 + 2];
      float c = cosf(0.5f * th), s = sinf(0.5f * th);
      float cp = cosf(0.5f * (phi + om)), sp = sinf(0.5f * (phi + om));
      float cm = cosf(0.5f * (phi - om)), sm = sinf(0.5f * (phi - om));
      sG[tid][0][0][0] =  cp * c;  sG[tid][0][0][1] = -sp * c;   // e^{-i(phi+om)/2} c
      sG[tid][0][1][0] = -cm * s;  sG[tid][0][1][1] = -sm * s;   // -e^{ i(phi-om)/2} s
      sG[tid][1][0][0] =  cm * s;  sG[tid][1][0][1] = -sm * s;   //  e^{-i(phi-om)/2} s
      sG[tid][1][1][0] =  cp * c;  sG[tid][1][1][1] =  sp * c;   //  e^{ i(phi+om)/2} c
    }
    __syncthreads();

    // ---- fuse 4 single-qubit gates per group into 16x16 complex U (Kronecker)
    {
      int j = tid >> 4, k = tid & 15;
      #pragma unroll
      for (int g = 0; g < 3; ++g) {
        float pr = 1.0f, pi = 0.0f;
        #pragma unroll
        for (int bit = 0; bit < 4; ++bit) {
          int q = 4 * g + bit;
          float gr = sG[q][(j >> bit) & 1][(k >> bit) & 1][0];
          float gi = sG[q][(j >> bit) & 1][(k >> bit) & 1][1];
          float nr = pr * gr - pi * gi;
          float ni = pr * gi + pi * gr;
          pr = nr; pi = ni;
        }
        sUr[g][j][k] = pr; sUi[g][j][k] = pi; sUiN[g][j][k] = -pi;
      }
    }
    __syncthreads();

    // ---- apply the 3 fused gates via V_WMMA_F32_16X16X4_F32 on LDS state ----
    const int lane  = tid & 31;
    const int wid   = tid >> 5;            // 8 waves
    const int mrow  = lane & 15;           // A:M / B:N / D:N index
    const int khalf = (lane >> 4) << 1;    // K sub-offset per lane half (0 or 2)

    for (int g = 0; g < 3; ++g) {
      // A fragments: U[M][K], 16x4 per k-tile, 2 VGPRs/lane (shared by all n-tiles)
      v2f aUr[4], aUi[4], aUiN[4];
      #pragma unroll
      for (int kt = 0; kt < 4; ++kt) {
        int kc = 4 * kt + khalf;
        aUr [kt].x = sUr [g][mrow][kc]; aUr [kt].y = sUr [g][mrow][kc + 1];
        aUi [kt].x = sUi [g][mrow][kc]; aUi [kt].y = sUi [g][mrow][kc + 1];
        aUiN[kt].x = sUiN[g][mrow][kc]; aUiN[kt].y = sUiN[g][mrow][kc + 1];
      }
      #pragma unroll
      for (int rep = 0; rep < 2; ++rep) {
        int nt  = wid + 8 * rep;           // 16 n-tiles over 8 waves
        int col = nt * 16 + mrow;
        // B fragments: S[K][col], mirrored A layout (4x16, K by vgpr+lanehalf)
        v2f bR[4], bI[4];
        #pragma unroll
        for (int kt = 0; kt < 4; ++kt) {
          int r0 = 4 * kt + khalf;
          int a0 = addr_of(g, r0,     col);
          int a1 = addr_of(g, r0 + 1, col);
          bR[kt].x = sRe[a0]; bR[kt].y = sRe[a1];
          bI[kt].x = sIm[a0]; bI[kt].y = sIm[a1];
        }
        // complex GEMM: Dr = Ur*Sr + (-Ui)*Si ; Di = Ur*Si + Ui*Sr
        v8f cr = {0,0,0,0,0,0,0,0};
        v8f ci = {0,0,0,0,0,0,0,0};
        #pragma unroll
        for (int kt = 0; kt < 4; ++kt) {
          cr = __builtin_amdgcn_wmma_f32_16x16x4_f32(false, aUr [kt], false, bR[kt], (short)0, cr, false, false);
          cr = __builtin_amdgcn_wmma_f32_16x16x4_f32(false, aUiN[kt], false, bI[kt], (short)0, cr, false, false);
          ci = __builtin_amdgcn_wmma_f32_16x16x4_f32(false, aUr [kt], false, bI[kt], (short)0, ci, false, false);
          ci = __builtin_amdgcn_wmma_f32_16x16x4_f32(false, aUi [kt], false, bR[kt], (short)0, ci, false, false);
        }
        // D layout: N = lane&15, M = v + 8*(lane>=16); same cols as inputs -> in-place OK
        #pragma unroll
        for (int v = 0; v < 8; ++v) {
          int M = v + ((lane >> 4) << 3);
          int a = addr_of(g, M, col);
          sRe[a] = cr[v]; sIm[a] = ci[v];
        }
      }
      __syncthreads();                     // cols remap between groups
    }

    // ---- CNOT ring, control i -> target (i+r)%12, r = l+1 ----
    int r = l + 1;
    for (int i = 0; i < NQ; ++i) {
      int c = i, tt = (i + r) % NQ;
      int lo = c < tt ? c : tt;
      int hi = c < tt ? tt : c;
      int mlo = (1 << lo) - 1, mhi = (1 << hi) - 1;
      for (int p = tid; p < 1024; p += NTHR) {
        int x = ((p & ~mlo) << 1) | (p & mlo);     // insert 0 at bit lo
        int s = ((x & ~mhi) << 1) | (x & mhi);     // insert 0 at bit hi
        s |= (1 << c);                             // control = 1, target = 0
        int s1 = s | (1 << tt);
        float tr = sRe[s]; sRe[s] = sRe[s1]; sRe[s1] = tr;
        float ti = sIm[s]; sIm[s] = sIm[s1]; sIm[s1] = ti;
      }
      __syncthreads();
    }
  }

  // ---- PauliZ expectations: z_q = sum_s |amp|^2 * (1 - 2*bit_q(s)) ----
  float acc[NQ];
  #pragma unroll
  for (int q = 0; q < NQ; ++q) acc[q] = 0.0f;
  for (int s = tid; s < NSTATE; s += NTHR) {
    float p = sRe[s] * sRe[s] + sIm[s] * sIm[s];
    #pragma unroll
    for (int q = 0; q < NQ; ++q) acc[q] += ((s >> q) & 1) ? -p : p;
  }
  #pragma unroll
  for (int q = 0; q < NQ; ++q) {
    float v = acc[q];
    for (int off = 16; off > 0; off >>= 1) v += __shfl_down(v, off, 32);
    acc[q] = v;
  }
  if ((tid & 31) == 0) {
    #pragma unroll
    for (int q = 0; q < NQ; ++q) sWred[tid >> 5][q] = acc[q];
  }
  __syncthreads();
  if (tid < NQ) {
    float z = 0.0f;
    #pragma unroll
    for (int w = 0; w < 8; ++w) z += sWred[w][tid];
    zbuf[(gate * BATCH + b) * NQ + tid] = z;
  }
}

__global__ __launch_bounds__(256) void qlstm_cell_kernel(
    const float* __restrict__ zbuf,
    const float* __restrict__ fW, const float* __restrict__ fb,
    const float* __restrict__ iW, const float* __restrict__ ib,
    const float* __restrict__ uW, const float* __restrict__ ub,
    const float* __restrict__ oW, const float* __restrict__ ob,
    float* __restrict__ hbuf, float* __restrict__ cbuf,
    float* __restrict__ out, int t)
{
  int idx = threadIdx.x;               // 0..255 = b*HID + h
  int b = idx >> 2, hh = idx & 3;
  const float* zF = zbuf + (0 * BATCH + b) * NQ;
  const float* zI = zbuf + (1 * BATCH + b) * NQ;
  const float* zU = zbuf + (2 * BATCH + b) * NQ;
  const float* zO = zbuf + (3 * BATCH + b) * NQ;
  float af = fb[hh], ai = ib[hh], au = ub[hh], ao = ob[hh];
  #pragma unroll
  for (int k = 0; k < NQ; ++k) {
    af += zF[k] * fW[hh * NQ + k];
    ai += zI[k] * iW[hh * NQ + k];
    au += zU[k] * uW[hh * NQ + k];
    ao += zO[k] * oW[hh * NQ + k];
  }
  float f = 1.0f / (1.0f + expf(-af));
  float i = 1.0f / (1.0f + expf(-ai));
  float g = tanhf(au);
  float o = 1.0f / (1.0f + expf(-ao));
  float c = f * cbuf[idx] + i * g;
  float h = o * tanhf(c);
  cbuf[idx] = c; hbuf[idx] = h;
  out[t * (BATCH * HID) + idx] = h;                          // outputs (S,B,H)
  out[SEQLEN * BATCH * HID + idx] = h;                       // hx
  out[SEQLEN * BATCH * HID + BATCH * HID + idx] = c;         // cx
}

__global__ void qlstm_init_kernel(float* hbuf, float* cbuf) {
  int i = threadIdx.x;
  if (i < BATCH * HID) { hbuf[i] = 0.0f; cbuf[i] = 0.0f; }
}

extern "C" void kernel_launch(void* const* d_in, const int* in_sizes, int n_in,
                              void* d_out, int out_size, void* d_ws, size_t ws_size,
                              hipStream_t stream) {
  (void)in_sizes; (void)n_in; (void)out_size; (void)ws_size;
  const float* inputs = (const float*)d_in[0];
  const float* wF = (const float*)d_in[1];
  const float* wI = (const float*)d_in[2];
  const float* wU = (const float*)d_in[3];
  const float* wO = (const float*)d_in[4];
  const float* fW = (const float*)d_in[5];
  const float* fb = (const float*)d_in[6];
  const float* iW = (const float*)d_in[7];
  const float* ib = (const float*)d_in[8];
  const float* uW = (const float*)d_in[9];
  const float* ub = (const float*)d_in[10];
  const float* oW = (const float*)d_in[11];
  const float* ob = (const float*)d_in[12];
  float* out  = (float*)d_out;
  float* zbuf = (float*)d_ws;                 // 4*64*12 floats
  float* hbuf = zbuf + 4 * BATCH * NQ;        // 256 floats
  float* cbuf = hbuf + BATCH * HID;           // 256 floats

  qlstm_init_kernel<<<1, 256, 0, stream>>>(hbuf, cbuf);
  for (int t = 0; t < SEQLEN; ++t) {
    qlstm_circuit_kernel<<<dim3(BATCH, 4), NTHR, 0, stream>>>(
        inputs, wF, wI, wU, wO, hbuf, zbuf, t);
    qlstm_cell_kernel<<<1, 256, 0, stream>>>(
        zbuf, fW, fb, iW, ib, uW, ub, oW, ob, hbuf, cbuf, out, t);
  }
}